// PoseGraphModule_42760694399067
// MI455X (gfx1250) — compile-verified
//
#include <hip/hip_runtime.h>
#include <math.h>

// ---------------------------------------------------------------------------
// Pose-graph edge error: error = se3_log( meas^-1 * (src^-1 * dst) ) * weights
// Memory-bound (~168 MB streamed @ 23.3 TB/s => ~7 us floor). Pose table is
// 5.6 MB -> L2 resident; stream everything else with non-temporal hints so
// the gathers keep hitting L2. One edge per lane, wave32, fp32, branchless.
// ---------------------------------------------------------------------------

#define NT_LOAD(p)     __builtin_nontemporal_load(p)
#define NT_STORE(v, p) __builtin_nontemporal_store((v), (p))

struct V3 { float x, y, z; };
struct Q4 { float x, y, z, w; };
struct Pose { V3 t; Q4 q; };

__device__ __forceinline__ V3 cross3(V3 a, V3 b) {
    return { a.y * b.z - a.z * b.y,
             a.z * b.x - a.x * b.z,
             a.x * b.y - a.y * b.x };
}
__device__ __forceinline__ float dot3(V3 a, V3 b) {
    return a.x * b.x + a.y * b.y + a.z * b.z;
}

// q * t rotation: u = 2*cross(v,t); t + w*u + cross(v,u)
__device__ __forceinline__ V3 qrot(Q4 q, V3 t) {
    V3 v{ q.x, q.y, q.z };
    V3 u = cross3(v, t);
    u.x *= 2.0f; u.y *= 2.0f; u.z *= 2.0f;
    V3 cu = cross3(v, u);
    return { t.x + q.w * u.x + cu.x,
             t.y + q.w * u.y + cu.y,
             t.z + q.w * u.z + cu.z };
}

__device__ __forceinline__ Q4 qmul(Q4 a, Q4 b) {
    V3 va{ a.x, a.y, a.z }, vb{ b.x, b.y, b.z };
    V3 c = cross3(va, vb);
    Q4 r;
    r.w = a.w * b.w - dot3(va, vb);
    r.x = a.w * b.x + b.w * a.x + c.x;
    r.y = a.w * b.y + b.w * a.y + c.y;
    r.z = a.w * b.z + b.w * a.z + c.z;
    return r;
}

__device__ __forceinline__ Pose se3_inv(Pose p) {
    Q4 qi{ -p.q.x, -p.q.y, -p.q.z, p.q.w };
    V3 ti = qrot(qi, p.t);
    Pose r;
    r.t = { -ti.x, -ti.y, -ti.z };
    r.q = qi;
    return r;
}

__device__ __forceinline__ Pose se3_mul(Pose a, Pose b) {
    Pose r;
    V3 rt = qrot(a.q, b.t);
    r.t = { a.t.x + rt.x, a.t.y + rt.y, a.t.z + rt.z };
    r.q = qmul(a.q, b.q);
    return r;
}

// Branchless pose gather: index 0 is the implicit identity pose, k>=1 maps to
// poses_rest[k-1]. Always issue the wide load from a clamped (valid) row, then
// select the identity with v_cndmask. Avoids exec-mask save/branch per gather;
// the pose table is L2-resident so the redundant row-0 load is a cheap hit.
__device__ __forceinline__ Pose load_pose(const float* __restrict__ pr, int idx) {
    int row = (idx > 0) ? (idx - 1) : 0;
    const float* r = pr + (size_t)row * 7;
    float f0 = r[0], f1 = r[1], f2 = r[2], f3 = r[3], f4 = r[4], f5 = r[5], f6 = r[6];
    bool valid = idx > 0;
    Pose p;
    p.t = { valid ? f0 : 0.0f, valid ? f1 : 0.0f, valid ? f2 : 0.0f };
    p.q = { valid ? f3 : 0.0f, valid ? f4 : 0.0f, valid ? f5 : 0.0f,
            valid ? f6 : 1.0f };
    return p;
}

__global__ __launch_bounds__(256) void pose_graph_error_kernel(
    const float* __restrict__ poses_rest,   // (NUM_POSES-1) x 7
    const float* __restrict__ edge_meas,    // NUM_EDGES x 7
    const float* __restrict__ edge_weights, // NUM_EDGES x 6
    const int*   __restrict__ edges_src,    // NUM_EDGES
    const int*   __restrict__ edges_dst,    // NUM_EDGES
    float*       __restrict__ out,          // NUM_EDGES x 6 (flat)
    int n_edges)
{
    int i = blockIdx.x * blockDim.x + threadIdx.x;
    if (i >= n_edges) return;

    // Indices: streamed once -> non-temporal.
    int si = NT_LOAD(edges_src + i);
    int di = NT_LOAD(edges_dst + i);

    // Measurement pose for this edge: streamed once -> non-temporal.
    const float* m = edge_meas + (size_t)i * 7;
    Pose meas;
    meas.t = { NT_LOAD(m + 0), NT_LOAD(m + 1), NT_LOAD(m + 2) };
    meas.q = { NT_LOAD(m + 3), NT_LOAD(m + 4), NT_LOAD(m + 5), NT_LOAD(m + 6) };

    Pose src = load_pose(poses_rest, si);
    Pose dst = load_pose(poses_rest, di);

    // diff = meas^-1 * (src^-1 * dst)
    Pose pred = se3_mul(se3_inv(src), dst);
    Pose diff = se3_mul(se3_inv(meas), pred);

    // ------------------------- se3_log(diff) -------------------------------
    Q4 q = diff.q;
    float qs = (q.w < 0.0f) ? -1.0f : 1.0f;   // select, no branch
    q.x *= qs; q.y *= qs; q.z *= qs; q.w *= qs;
    V3 v{ q.x, q.y, q.z };

    float n2      = dot3(v, v);
    bool  small_n = n2 < 1e-12f;
    float n_safe  = sqrtf(small_n ? 1.0f : n2);
    float theta_q = 2.0f * atan2f(sqrtf(n2), q.w);
    float scale   = small_n ? (2.0f / q.w) : (theta_q / n_safe);

    V3 phi{ scale * v.x, scale * v.y, scale * v.z };

    float th2  = dot3(phi, phi);
    bool  smll = th2 < 1e-8f;
    float th2s = smll ? 1.0f : th2;          // guard so both select arms are finite
    float ths  = sqrtf(th2s);

    float coef_small = (1.0f / 12.0f) + th2 * (1.0f / 720.0f);
    float sn, cn;
    sincosf(ths, &sn, &cn);                  // one shared range reduction
    float coef_big = 1.0f / th2s - (1.0f + cn) / (2.0f * ths * sn);
    float coef = smll ? coef_small : coef_big;

    V3 pxt  = cross3(phi, diff.t);
    V3 ppxt = cross3(phi, pxt);
    V3 rho{ diff.t.x - 0.5f * pxt.x + coef * ppxt.x,
            diff.t.y - 0.5f * pxt.y + coef * ppxt.y,
            diff.t.z - 0.5f * pxt.z + coef * ppxt.z };

    // Weighted output: streamed once -> non-temporal load + store.
    const float* wp = edge_weights + (size_t)i * 6;
    float*       op = out          + (size_t)i * 6;
    NT_STORE(rho.x * NT_LOAD(wp + 0), op + 0);
    NT_STORE(rho.y * NT_LOAD(wp + 1), op + 1);
    NT_STORE(rho.z * NT_LOAD(wp + 2), op + 2);
    NT_STORE(phi.x * NT_LOAD(wp + 3), op + 3);
    NT_STORE(phi.y * NT_LOAD(wp + 4), op + 4);
    NT_STORE(phi.z * NT_LOAD(wp + 5), op + 5);
}

extern "C" void kernel_launch(void* const* d_in, const int* in_sizes, int n_in,
                              void* d_out, int out_size, void* d_ws, size_t ws_size,
                              hipStream_t stream) {
    // setup_inputs() order:
    //   0: poses_rest   (NUM_POSES-1, 7)  float32
    //   1: edge_meas    (NUM_EDGES, 7)    float32
    //   2: edge_weights (NUM_EDGES, 6)    float32
    //   3: edges_src    (NUM_EDGES,)      int32
    //   4: edges_dst    (NUM_EDGES,)      int32
    const float* poses_rest   = (const float*)d_in[0];
    const float* edge_meas    = (const float*)d_in[1];
    const float* edge_weights = (const float*)d_in[2];
    const int*   edges_src    = (const int*)d_in[3];
    const int*   edges_dst    = (const int*)d_in[4];
    float*       out          = (float*)d_out;

    int n_edges = in_sizes[3];

    const int block = 256; // 8 wave32s per block
    const int grid  = (n_edges + block - 1) / block;
    pose_graph_error_kernel<<<grid, block, 0, stream>>>(
        poses_rest, edge_meas, edge_weights, edges_src, edges_dst, out, n_edges);
}